// AugementModel_25623774888353
// MI455X (gfx1250) — compile-verified
//
#include <hip/hip_runtime.h>
#include <stdint.h>

#define TILE 256   // rows per block == threads per block (8 waves, wave32)
#define ROWF 10    // floats per state/output row

// Physical constants folded at compile time:
//   1/(R*TAU) = 1/(0.05*5) = 4,  1/TAU = 0.2,  DELTA_P/TAU = 0.02
#define C_M      10.0f
#define C_D      1.0f
#define C_IRTAU  4.0f
#define C_ITAU   0.2f
#define C_DPTAU  0.02f

typedef __attribute__((ext_vector_type(4))) float v4f;  // true clang vector (not HIP_vector_type)

__global__ __launch_bounds__(TILE) void
aug_sens_kernel(const float* __restrict__ state,
                const float* __restrict__ Kp,
                float* __restrict__ out,
                int B)
{
    __shared__ __align__(16) float sh[TILE * ROWF];   // 10,240 B staging tile

    const int t = threadIdx.x;
    const long long blockStart = (long long)blockIdx.x * TILE;
    int validRows = B - (int)blockStart;
    if (validRows > TILE) validRows = TILE;
    const bool fullTile = (validRows == TILE);        // uniform per block

    const uint32_t shBase = (uint32_t)(uintptr_t)sh;  // low 32 bits of flat shared ptr == LDS byte offset
    const char* src = (const char*)(state + blockStart * (long long)ROWF);
    const char* dst = (const char*)(out   + blockStart * (long long)ROWF);

    // ---- Stage 1: async DMA global -> LDS, fully coalesced (8 B/lane, 256 B/wave/issue)
    if (fullTile) {
        // No guards: straight run of 5 async issues, no exec-mask manipulation.
#pragma unroll
        for (int k = 0; k < 5; ++k) {
            const uint32_t off = (uint32_t)((t + k * TILE) * 8);
            asm volatile("global_load_async_to_lds_b64 %0, %1, %2 th:TH_LOAD_NT"
                         :: "v"(shBase + off), "v"(off), "s"(src)
                         : "memory");
        }
    } else {
        const int validF2 = (validRows * ROWF) >> 1;  // row size is even in floats
#pragma unroll
        for (int k = 0; k < 5; ++k) {
            const int f2i = t + k * TILE;
            if (f2i < validF2) {
                const uint32_t off = (uint32_t)(f2i * 8);
                asm volatile("global_load_async_to_lds_b64 %0, %1, %2"
                             :: "v"(shBase + off), "v"(off), "s"(src)
                             : "memory");
            }
        }
    }
    asm volatile("s_wait_asynccnt 0" ::: "memory");
    __syncthreads();

    // ---- Stage 2: per-row compute (thread t owns row t; LDS row accesses are bank-conflict-free:
    //      dword addresses 10*t are distinct mod 64 across the 32 lanes of a wave)
    if (t < validRows) {
        float* row = &sh[t * ROWF];
        const float w  = row[0];
        const float wd = row[1];

        // K row is 16 B aligned, stride 16 -> naturally coalesced b128; read-once -> non-temporal
        const v4f kk = __builtin_nontemporal_load(
            (const v4f*)(Kp + 4ll * (blockStart + t)));
        const float k0 = kk.x, k1 = kk.y, k2 = kk.z, k3 = kk.w;

        const float M    = C_M - k0 * w - k1 * wd;
        const float D    = C_D - k2 * w - k3 * wd;
        const float invM = 1.0f / M;
        const float A    = (C_IRTAU + D * C_ITAU) * invM;  // 1/(R*TAU*M) + D/(TAU*M)
        const float Bc   = D * invM + C_ITAU;              // D/M + 1/TAU
        const float f2   = -A * w - Bc * wd + C_DPTAU * invM;

        row[0] = wd;     // primal: d(omega)/dt
        row[1] = f2;     // primal: d(omega_dot)/dt

#pragma unroll
        for (int p = 0; p < 4; ++p) {
            const float tw  = row[2 + 2 * p];
            const float twd = row[3 + 2 * p];
            float dM = -(k0 * tw + k1 * twd);
            float dD = -(k2 * tw + k3 * twd);
            if      (p == 0) dM -= w;
            else if (p == 1) dM -= wd;
            else if (p == 2) dD -= w;
            else             dD -= wd;
            const float dinvM = -invM * invM * dM;
            const float dA    = (dD * C_ITAU) * invM + (C_IRTAU + D * C_ITAU) * dinvM;
            const float dBc   = dD * invM + D * dinvM;
            const float df2   = -(dA * w + A * tw) - (dBc * wd + Bc * twd) + C_DPTAU * dinvM;
            row[2 + 2 * p] = twd;   // d f1 / tangent p
            row[3 + 2 * p] = df2;   // d f2 / tangent p
        }
    }
    __syncthreads();

    // ---- Stage 3: async DMA LDS -> global, fully coalesced, non-temporal (write-once stream)
    if (fullTile) {
#pragma unroll
        for (int k = 0; k < 5; ++k) {
            const uint32_t off = (uint32_t)((t + k * TILE) * 8);
            asm volatile("global_store_async_from_lds_b64 %0, %1, %2 th:TH_STORE_NT"
                         :: "v"(off), "v"(shBase + off), "s"(dst)
                         : "memory");
        }
    } else {
        const int validF2 = (validRows * ROWF) >> 1;
#pragma unroll
        for (int k = 0; k < 5; ++k) {
            const int f2i = t + k * TILE;
            if (f2i < validF2) {
                const uint32_t off = (uint32_t)(f2i * 8);
                asm volatile("global_store_async_from_lds_b64 %0, %1, %2"
                             :: "v"(off), "v"(shBase + off), "s"(dst)
                             : "memory");
            }
        }
    }
    asm volatile("s_wait_asynccnt 0" ::: "memory");
}

extern "C" void kernel_launch(void* const* d_in, const int* in_sizes, int n_in,
                              void* d_out, int out_size, void* d_ws, size_t ws_size,
                              hipStream_t stream)
{
    // setup_inputs order: {'t': scalar, 'state': (B, 10) f32, 'K': (B, 4) f32}
    const float* state = (const float*)d_in[1];
    const float* Kp    = (const float*)d_in[2];
    float* out         = (float*)d_out;

    const int B = in_sizes[2] / 4;               // K is (B, 4)
    const int blocks = (B + TILE - 1) / TILE;    // 15,625 blocks for B = 4e6

    aug_sens_kernel<<<blocks, TILE, 0, stream>>>(state, Kp, out, B);
}